// GraphConv_15212774163211
// MI455X (gfx1250) — compile-verified
//
#include <hip/hip_runtime.h>
#include <math.h>

#define N_USERS    50000
#define N_ITEMS    30000
#define N_ENTITIES 150000
#define N_ASPECTS  32
#define CHANNEL    64
#define N_EDGES    2000000

typedef __attribute__((ext_vector_type(2))) float v2f;
typedef __attribute__((ext_vector_type(8))) float v8f;

// ---------------- generic helpers ----------------

__global__ void zero_f32(float* __restrict__ p, int n) {
    int i = blockIdx.x * blockDim.x + threadIdx.x;
    if (i < n) p[i] = 0.0f;
}

// entity_cur = entity_emb; out_item (first N_ITEMS rows) = entity_emb[:N_ITEMS]
__global__ void init_entity(const float* __restrict__ entity_emb,
                            float* __restrict__ entity_cur,
                            float* __restrict__ out_item, int n_total, int n_item_elems) {
    int i = blockIdx.x * blockDim.x + threadIdx.x;
    if (i < n_total) {
        float v = entity_emb[i];
        entity_cur[i] = v;
        if (i < n_item_elems) out_item[i] = v;
    }
}

// cnt[head[e]] += 1  (hop-invariant)
__global__ void edge_count(const int* __restrict__ head, float* __restrict__ cnt, int ne) {
    int e = blockIdx.x * blockDim.x + threadIdx.x;
    if (e < ne) atomicAdd(&cnt[head[e]], 1.0f);
}

// one wave per edge, 2 channels per lane:
// sums[head] += entity_cur[tail] * weight[etype-2]
__global__ void edge_scatter(const int* __restrict__ head, const int* __restrict__ tail,
                             const int* __restrict__ etype,
                             const float* __restrict__ weight,
                             const float* __restrict__ entity_cur,
                             float* __restrict__ sums, int ne) {
    int g = blockIdx.x * blockDim.x + threadIdx.x;
    int e = g >> 5;
    int lane = (g & 31) * 2;
    if (e >= ne) return;
    int h = head[e];
    int t = tail[e];
    int r = etype[e] - 2;
    const float* ep = entity_cur + (size_t)t * CHANNEL + lane;
    const float* wp = weight + (size_t)r * CHANNEL + lane;
    float v0 = ep[0] * wp[0];
    float v1 = ep[1] * wp[1];
    float* sp = sums + (size_t)h * CHANNEL + lane;
    atomicAdd(sp + 0, v0);
    atomicAdd(sp + 1, v1);
}

// entity_cur[row] = l2norm(sums[row] / max(cnt,1)); out_item[row] += that (row < N_ITEMS)
// one wave (32 lanes x 2 channels) per row, 8 rows per 256-thread block
__global__ void entity_norm_acc(const float* __restrict__ sums,
                                const float* __restrict__ cnt,
                                float* __restrict__ entity_cur,
                                float* __restrict__ out_item, int nrows, int nitems) {
    int wave = threadIdx.x >> 5;
    int lane = threadIdx.x & 31;
    int row = blockIdx.x * 8 + wave;
    if (row >= nrows) return;
    size_t off = (size_t)row * CHANNEL + lane * 2;
    float c = fmaxf(cnt[row], 1.0f);
    float x0 = sums[off] / c;
    float x1 = sums[off + 1] / c;
    float ss = x0 * x0 + x1 * x1;
    #pragma unroll
    for (int m = 16; m >= 1; m >>= 1) ss += __shfl_xor(ss, m, 32);
    float inv = 1.0f / fmaxf(sqrtf(ss), 1e-12f);
    x0 *= inv; x1 *= inv;
    entity_cur[off]     = x0;
    entity_cur[off + 1] = x1;
    if (row < nitems) {
        out_item[off]     += x0;
        out_item[off + 1] += x1;
    }
}

// ---------------- WMMA GEMM: C[M x 64] = A[M x 32] * B[32 x 64], f32 ----------------
// One wave per 16-row M tile, full N=64 (4 accumulators). M must be a multiple of 16
// (50000 and 30000 both are). Guard is wave-uniform so EXEC stays all-ones for WMMA.
__global__ void gemm_a32xb64_wmma(const float* __restrict__ A,
                                  const float* __restrict__ B,
                                  float* __restrict__ C, int mtiles) {
    int wave = threadIdx.x >> 5;
    int lane = threadIdx.x & 31;
    int mtile = blockIdx.x * 8 + wave;
    if (mtile >= mtiles) return;
    int half = lane >> 4;   // 0: K pair {k,k+1}, 1: K pair {k+2,k+3}
    int l16  = lane & 15;
    int mbase = mtile * 16;

    v8f acc0 = {}; v8f acc1 = {}; v8f acc2 = {}; v8f acc3 = {};
    const float* arow = A + (size_t)(mbase + l16) * N_ASPECTS + half * 2;

    #pragma unroll
    for (int k = 0; k < N_ASPECTS; k += 4) {
        v2f a;
        a.x = arow[k];
        a.y = arow[k + 1];
        const float* bp = B + (size_t)(k + half * 2) * CHANNEL + l16;
        v2f b0, b1, b2, b3;
        b0.x = bp[0];  b0.y = bp[0 + CHANNEL];
        b1.x = bp[16]; b1.y = bp[16 + CHANNEL];
        b2.x = bp[32]; b2.y = bp[32 + CHANNEL];
        b3.x = bp[48]; b3.y = bp[48 + CHANNEL];
        acc0 = __builtin_amdgcn_wmma_f32_16x16x4_f32(false, a, false, b0, (short)0, acc0, false, false);
        acc1 = __builtin_amdgcn_wmma_f32_16x16x4_f32(false, a, false, b1, (short)0, acc1, false, false);
        acc2 = __builtin_amdgcn_wmma_f32_16x16x4_f32(false, a, false, b2, (short)0, acc2, false, false);
        acc3 = __builtin_amdgcn_wmma_f32_16x16x4_f32(false, a, false, b3, (short)0, acc3, false, false);
    }
    #pragma unroll
    for (int v = 0; v < 8; ++v) {
        int row = mbase + v + half * 8;
        float* crow = C + (size_t)row * CHANNEL + l16;
        crow[0]  = acc0[v];
        crow[16] = acc1[v];
        crow[32] = acc2[v];
        crow[48] = acc3[v];
    }
}

// out_user[row] = user_emb[row] + 3 * l2norm(U[row])
__global__ void user_epilogue(const float* __restrict__ U,
                              const float* __restrict__ user_emb,
                              float* __restrict__ out_user, int nrows) {
    int wave = threadIdx.x >> 5;
    int lane = threadIdx.x & 31;
    int row = blockIdx.x * 8 + wave;
    if (row >= nrows) return;
    size_t off = (size_t)row * CHANNEL + lane * 2;
    float x0 = U[off], x1 = U[off + 1];
    float ss = x0 * x0 + x1 * x1;
    #pragma unroll
    for (int m = 16; m >= 1; m >>= 1) ss += __shfl_xor(ss, m, 32);
    float inv = 1.0f / fmaxf(sqrtf(ss), 1e-12f);
    out_user[off]     = user_emb[off]     + 3.0f * x0 * inv;
    out_user[off + 1] = user_emb[off + 1] + 3.0f * x1 * inv;
}

// out_item[row] = W1 * out_item[row](=entity_res) + W2 * (item_emb[row] + 3*l2norm(I[row]))
__global__ void item_epilogue(const float* __restrict__ I,
                              const float* __restrict__ item_emb,
                              const float* __restrict__ W1p,
                              const float* __restrict__ W2p,
                              float* __restrict__ out_item, int nrows) {
    int wave = threadIdx.x >> 5;
    int lane = threadIdx.x & 31;
    int row = blockIdx.x * 8 + wave;
    if (row >= nrows) return;
    float w1 = W1p[0], w2 = W2p[0];
    size_t off = (size_t)row * CHANNEL + lane * 2;
    float x0 = I[off], x1 = I[off + 1];
    float ss = x0 * x0 + x1 * x1;
    #pragma unroll
    for (int m = 16; m >= 1; m >>= 1) ss += __shfl_xor(ss, m, 32);
    float inv = 1.0f / fmaxf(sqrtf(ss), 1e-12f);
    float e0 = out_item[off], e1 = out_item[off + 1];
    out_item[off]     = w1 * e0 + w2 * (item_emb[off]     + 3.0f * x0 * inv);
    out_item[off + 1] = w1 * e1 + w2 * (item_emb[off + 1] + 3.0f * x1 * inv);
}

extern "C" void kernel_launch(void* const* d_in, const int* in_sizes, int n_in,
                              void* d_out, int out_size, void* d_ws, size_t ws_size,
                              hipStream_t stream) {
    const float* user_emb   = (const float*)d_in[0];
    const float* item_emb   = (const float*)d_in[1];
    const float* entity_emb = (const float*)d_in[2];
    const float* aspect_emb = (const float*)d_in[3];
    const float* ua_mat     = (const float*)d_in[4];
    const float* ia_mat     = (const float*)d_in[5];
    const float* weight     = (const float*)d_in[6];
    const float* W1         = (const float*)d_in[7];
    const float* W2         = (const float*)d_in[8];
    const int*   edge_index = (const int*)d_in[9];
    const int*   edge_type  = (const int*)d_in[10];
    const int* head = edge_index;
    const int* tail = edge_index + N_EDGES;

    float* out_item = (float*)d_out;                           // [30000 x 64]
    float* out_user = (float*)d_out + (size_t)N_ITEMS * CHANNEL; // [50000 x 64]

    char* w = (char*)d_ws;
    float* entity_cur = (float*)w; w += (size_t)N_ENTITIES * CHANNEL * sizeof(float);
    float* sums       = (float*)w; w += (size_t)N_ENTITIES * CHANNEL * sizeof(float);
    float* cnt        = (float*)w; w += (size_t)N_ENTITIES * sizeof(float);
    float* Ubuf       = (float*)w; w += (size_t)N_USERS * CHANNEL * sizeof(float);
    float* Ibuf       = (float*)w; w += (size_t)N_ITEMS * CHANNEL * sizeof(float);

    const int B = 256;
    const int n_ent_elems  = N_ENTITIES * CHANNEL;   // 9.6M
    const int n_item_elems = N_ITEMS * CHANNEL;      // 1.92M

    // init entity_cur and the entity_res accumulator (= out_item region)
    init_entity<<<(n_ent_elems + B - 1) / B, B, 0, stream>>>(
        entity_emb, entity_cur, out_item, n_ent_elems, n_item_elems);

    // hop-invariant edge counts
    zero_f32<<<(N_ENTITIES + B - 1) / B, B, 0, stream>>>(cnt, N_ENTITIES);
    edge_count<<<(N_EDGES + B - 1) / B, B, 0, stream>>>(head, cnt, N_EDGES);

    // hop-invariant aspect GEMMs (WMMA) + user output
    gemm_a32xb64_wmma<<<(N_USERS / 16 + 7) / 8, B, 0, stream>>>(ua_mat, aspect_emb, Ubuf, N_USERS / 16);
    gemm_a32xb64_wmma<<<(N_ITEMS / 16 + 7) / 8, B, 0, stream>>>(ia_mat, aspect_emb, Ibuf, N_ITEMS / 16);
    user_epilogue<<<(N_USERS + 7) / 8, B, 0, stream>>>(Ubuf, user_emb, out_user, N_USERS);

    // 3 hops of KG gather-scatter-mean + normalize + residual accumulate
    for (int hop = 0; hop < 3; ++hop) {
        zero_f32<<<(n_ent_elems + B - 1) / B, B, 0, stream>>>(sums, n_ent_elems);
        long long threads = (long long)N_EDGES * 32;
        edge_scatter<<<(int)((threads + B - 1) / B), B, 0, stream>>>(
            head, tail, edge_type, weight, entity_cur, sums, N_EDGES);
        entity_norm_acc<<<(N_ENTITIES + 7) / 8, B, 0, stream>>>(
            sums, cnt, entity_cur, out_item, N_ENTITIES, N_ITEMS);
    }

    // final item combine
    item_epilogue<<<(N_ITEMS + 7) / 8, B, 0, stream>>>(Ibuf, item_emb, W1, W2, out_item, N_ITEMS);
}